// StatDenoiser_4509715661487
// MI455X (gfx1250) — compile-verified
//
#include <hip/hip_runtime.h>
#include <stdint.h>

#define RADIUS 5
#define KW     11
#define TILE   16
#define HALO   26            // TILE + 2*RADIUS
#define HH     256
#define WW     256
#define HP     266           // HH + 2*RADIUS (reflect-padded)
#define NPIX   (HP * HP)     // 70756
#define PLANE  (HH * WW)     // 65536
#define G2W    8.47376436796f          // GAMMA_W^2
#define NEGHL2E (-0.7213475204444817f) // -0.5 * log2(e)
#define RSTRIDE 33                     // padded Gram row stride (bank-friendly)

typedef unsigned int v4u __attribute__((ext_vector_type(4)));
typedef int          v8i __attribute__((ext_vector_type(8)));
typedef int          v4i __attribute__((ext_vector_type(4)));
typedef float        v2f __attribute__((ext_vector_type(2)));
typedef float        v8f __attribute__((ext_vector_type(8)));

__device__ __forceinline__ float fast_exp2(float x) {
#if __has_builtin(__builtin_amdgcn_exp2f)
    return __builtin_amdgcn_exp2f(x);   // v_exp_f32
#else
    return exp2f(x);
#endif
}

__device__ __forceinline__ int reflect_idx(int i, int n) {
    i = (i < 0) ? -i : i;
    return (i >= n) ? (2 * n - 2 - i) : i;
}

// Branchless per-channel t-test:  pass <=> d^2 < g^2*(vi+vj)
//   | pass if d==0 and s==0 (w=0.5 case) ; fail if min(vi,vj)==0 and d!=0.
__device__ __forceinline__ int ttest_pass(float d, float s, float vi, float vj) {
    const int a = (d * d) < (G2W * s);
    const int b = (d == 0.f) & (s == 0.f);
    const int z = (fminf(vi, vj) == 0.f) & (d != 0.f);
    return (a | b) & (z ^ 1);
}

// ---------------------------------------------------------------------------
// Stage 1: reflect-padded, pre-scaled, float4-packed planes in ws.
//   wsG : [NPIX][8]  ghat = guidance * (1/sqrt(sigma_c))
//   wsEV: [NPIX][8]  (e0,e1,e2,v0, v1,v2,|ghat|^2,0)   <- norm rides free slot
//   wsI : [NPIX][4]  (i0,i1,i2,1.0)
// ---------------------------------------------------------------------------
__global__ __launch_bounds__(256) void pack_pad(
    const float* __restrict__ image, const float* __restrict__ guidance,
    const float* __restrict__ est,   const float* __restrict__ var,
    float* __restrict__ wsG, float* __restrict__ wsEV, float* __restrict__ wsI)
{
    int p = blockIdx.x * 256 + threadIdx.x;
    if (p >= NPIX) return;
    int py = p / HP, px = p - py * HP;
    int s = reflect_idx(py - RADIUS, HH) * WW + reflect_idx(px - RADIUS, WW);

    const float isgA = 0.31622776601683794f; // 1/sqrt(10)
    const float isgB = 7.0710678118654755f;  // 1/sqrt(0.02)
    const float isgC = 3.1622776601683795f;  // 1/sqrt(0.1)

    float4 g0, g1;
    g0.x = guidance[0 * PLANE + s] * isgA;
    g0.y = guidance[1 * PLANE + s] * isgA;
    g0.z = guidance[2 * PLANE + s] * isgB;
    g0.w = guidance[3 * PLANE + s] * isgB;
    g1.x = guidance[4 * PLANE + s] * isgB;
    g1.y = guidance[5 * PLANE + s] * isgC;
    g1.z = guidance[6 * PLANE + s] * isgC;
    g1.w = guidance[7 * PLANE + s] * isgC;
    ((float4*)wsG)[2 * p + 0] = g0;
    ((float4*)wsG)[2 * p + 1] = g1;

    float nrm;
    nrm = g0.x * g0.x;
    nrm = fmaf(g0.y, g0.y, nrm);
    nrm = fmaf(g0.z, g0.z, nrm);
    nrm = fmaf(g0.w, g0.w, nrm);
    nrm = fmaf(g1.x, g1.x, nrm);
    nrm = fmaf(g1.y, g1.y, nrm);
    nrm = fmaf(g1.z, g1.z, nrm);
    nrm = fmaf(g1.w, g1.w, nrm);

    float4 ev0, ev1;
    ev0.x = est[0 * PLANE + s]; ev0.y = est[1 * PLANE + s]; ev0.z = est[2 * PLANE + s];
    ev0.w = var[0 * PLANE + s];
    ev1.x = var[1 * PLANE + s]; ev1.y = var[2 * PLANE + s];
    ev1.z = nrm; ev1.w = 0.f;
    ((float4*)wsEV)[2 * p + 0] = ev0;
    ((float4*)wsEV)[2 * p + 1] = ev1;

    float4 i4;
    i4.x = image[0 * PLANE + s]; i4.y = image[1 * PLANE + s]; i4.z = image[2 * PLANE + s];
    i4.w = 1.0f;
    ((float4*)wsI)[p] = i4;
}

// ---------------------------------------------------------------------------
// TDM descriptor issue: one 2D strided tile (13 rows x 26*C elements) per wave.
// ---------------------------------------------------------------------------
__device__ __forceinline__ void tdm_tile_load(unsigned lds_byte_off,
                                              unsigned long long gaddr,
                                              unsigned C /*floats per pixel*/)
{
    const unsigned td0   = HP * C;      // tensor_dim0 (elements)
    const unsigned td1   = HP;          // tensor_dim1 (rows)
    const unsigned tile0 = HALO * C;    // tile_dim0
    const unsigned tile1 = 13;          // tile_dim1
    const unsigned str0  = HP * C;      // tensor_dim0_stride (elements)

    v4u g0;
    g0[0] = 1u;                                                  // count=1
    g0[1] = lds_byte_off;                                        // lds_addr
    g0[2] = (unsigned)gaddr;                                     // addr[31:0]
    g0[3] = (unsigned)((gaddr >> 32) & 0x1FFFFFFu) | (2u << 30); // addr[56:32]|type=2

    v8i g1;
    g1[0] = (int)(2u << 16);                               // data_size=2 (4B)
    g1[1] = (int)((td0 & 0xFFFFu) << 16);                  // tensor_dim0[15:0]
    g1[2] = (int)((td0 >> 16) | ((td1 & 0xFFFFu) << 16));  // dim0 hi | dim1 lo
    g1[3] = (int)((td1 >> 16) | (tile0 << 16));            // dim1 hi | tile_dim0
    g1[4] = (int)tile1;                                    // tile_dim1, tile_dim2=0
    g1[5] = (int)str0;                                     // stride0[31:0]
    g1[6] = 0;
    g1[7] = 0;

    v4i z4 = {0, 0, 0, 0};
#if __has_include(<hip/amd_detail/amd_gfx1250_TDM.h>)
    v8i z8 = {0, 0, 0, 0, 0, 0, 0, 0};
    __builtin_amdgcn_tensor_load_to_lds(g0, g1, z4, z4, z8, 0);
#else
    __builtin_amdgcn_tensor_load_to_lds(g0, g1, z4, z4, 0);
#endif
}

// ---------------------------------------------------------------------------
// Stage 2: one 16x16 tile per workgroup. TDM stages halo tile -> LDS,
// WMMA computes the guidance Gram blocks, VALU does the t-test + blend.
// ---------------------------------------------------------------------------
__global__ __launch_bounds__(256) void denoise_tiled(
    const float* __restrict__ wsG, const float* __restrict__ wsEV,
    const float* __restrict__ wsI, float* __restrict__ out)
{
    __shared__ float4 sG [HALO * HALO * 2];   // 21632 B  ghat, 8 ch AoS
    __shared__ float4 sEV[HALO * HALO * 2];   // 21632 B  est/var/norm
    __shared__ float4 sI [HALO * HALO];       // 10816 B  img + 1.0
    __shared__ float  sR [8 * 32 * RSTRIDE];  // 33792 B  per-wave Gram scratch

    const int tid = threadIdx.x;
    const int bx = blockIdx.x, by = blockIdx.y;

    // --- async tile staging: 6 TDM descriptors = {G,EV,I} x {top,bottom} ---
    const int wave = __builtin_amdgcn_readfirstlane(tid >> 5);   // scalar
    if (wave < 6) {
        const int a = wave >> 1;
        const int h = wave & 1;
        const unsigned C = (a == 2) ? 4u : 8u;
        const float* base = (a == 0) ? wsG : (a == 1) ? wsEV : wsI;
        unsigned lds0 = (a == 0) ? (unsigned)(uintptr_t)sG
                      : (a == 1) ? (unsigned)(uintptr_t)sEV
                                 : (unsigned)(uintptr_t)sI;
        const int hy0 = h * 13;
        unsigned long long gaddr = (unsigned long long)(uintptr_t)base +
            (unsigned long long)(((by * TILE + hy0) * HP + bx * TILE) * C) * 4ull;
        tdm_tile_load(lds0 + (unsigned)(hy0 * HALO * C * 4), gaddr, C);
    }
    __builtin_amdgcn_s_wait_tensorcnt(0);
    __syncthreads();

    const int ty = tid >> 4, tx = tid & 15;
    const int lane = tid & 31;
    const int half = lane >> 4;            // lane half selects K offset (+0/+2)
    const float* sGf = (const float*)sG;

    // --- A matrices (this wave's 32 tile pixels), loaded once ---------------
    // A 16x4 f32 layout: VGPR0/1 = (K, K+1), lanes16-31 use K+2 (ISA 7.12.2).
    v2f A[2][2];
#pragma unroll
    for (int mt = 0; mt < 2; ++mt) {
        const int pix  = (wave << 5) + (mt << 4) + (lane & 15);
        const int hidx = ((pix >> 4) + RADIUS) * HALO + (pix & 15) + RADIUS;
#pragma unroll
        for (int kc = 0; kc < 2; ++kc) {
            const int ch = (half << 1) + (kc << 2);
            A[mt][kc] = *(const v2f*)&sGf[hidx * 8 + ch];
        }
    }

    const int ci = (ty + RADIUS) * HALO + (tx + RADIUS);
    const float4 cev0 = sEV[2 * ci + 0], cev1 = sEV[2 * ci + 1];
    const float  np   = cev1.z;           // |ghat_p|^2

    float* Rw = &sR[wave * 32 * RSTRIDE];
    float accx = 0.f, accy = 0.f, accz = 0.f, accw = 0.f;

    // Wave w only ever needs halo rows hy in [2w, 2w+11] -> 12 iterations.
    for (int i = 0; i < 12; ++i) {
        const int hy = (wave << 1) + i;   // scalar per wave

        // --- B matrices: halo row hy, cols 16nt..16nt+15 --------------------
        v2f B[2][2];
#pragma unroll
        for (int nt = 0; nt < 2; ++nt) {
            int hidx = hy * HALO + (nt << 4) + (lane & 15);
            if (hidx > HALO * HALO - 1) hidx = HALO * HALO - 1;  // pad cols 26..31
#pragma unroll
            for (int kc = 0; kc < 2; ++kc) {
                const int ch = (half << 1) + (kc << 2);
                B[nt][kc] = *(const v2f*)&sGf[hidx * 8 + ch];
            }
        }

        // --- Gram block via V_WMMA_F32_16X16X4_F32 (K=8 chained) ------------
#pragma unroll
        for (int mt = 0; mt < 2; ++mt) {
#pragma unroll
            for (int nt = 0; nt < 2; ++nt) {
                v8f c = {0.f, 0.f, 0.f, 0.f, 0.f, 0.f, 0.f, 0.f};
                c = __builtin_amdgcn_wmma_f32_16x16x4_f32(
                        false, A[mt][0], false, B[nt][0], (short)0, c, false, false);
                c = __builtin_amdgcn_wmma_f32_16x16x4_f32(
                        false, A[mt][1], false, B[nt][1], (short)0, c, false, false);
                // D layout: VGPR r = (M=r+8*half, N=lane&15); scatter to scratch
#pragma unroll
                for (int r = 0; r < 8; ++r) {
                    const int m = (mt << 4) + r + (half << 3);
                    const int n = (nt << 4) + (lane & 15);
                    Rw[n * RSTRIDE + m] = c[r];
                }
            }
        }

        // --- consume: same wave wrote Rw, LDS is in-order per wave ----------
        const int dy = hy - ty;
        if (dy >= 0 && dy < KW) {
            const int rowb = hy * HALO + tx;
#pragma unroll
            for (int dx = 0; dx < KW; ++dx) {
                const int q = rowb + dx;
                const float4 ev0 = sEV[2 * q + 0], ev1 = sEV[2 * q + 1];
                const float gram = Rw[(tx + dx) * RSTRIDE + lane];
                const float maha = fmaf(-2.f, gram, np + ev1.z);
                const float bil  = fast_exp2(NEGHL2E * maha);

                // fully branchless 3-channel t-test
                const int ok = ttest_pass(cev0.x - ev0.x, cev0.w + ev0.w, cev0.w, ev0.w)
                             & ttest_pass(cev0.y - ev0.y, cev1.x + ev1.x, cev1.x, ev1.x)
                             & ttest_pass(cev0.z - ev0.z, cev1.y + ev1.y, cev1.y, ev1.y);
                const float wgt = ok ? bil : 0.f;

                const float4 i4 = sI[q];
                accx = fmaf(wgt, i4.x, accx);
                accy = fmaf(wgt, i4.y, accy);
                accz = fmaf(wgt, i4.z, accz);
                accw += wgt;
            }
        }
    }

    const float inv = 1.0f / fmaxf(accw, 1e-10f);
    const int o = (by * TILE + ty) * WW + (bx * TILE + tx);
    out[o]             = accx * inv;
    out[o + PLANE]     = accy * inv;
    out[o + 2 * PLANE] = accz * inv;
}

// ---------------------------------------------------------------------------
// Fallback (only if ws_size too small): same math, direct global loads.
// ---------------------------------------------------------------------------
__global__ __launch_bounds__(256) void denoise_direct(
    const float* __restrict__ image, const float* __restrict__ guidance,
    const float* __restrict__ est,   const float* __restrict__ var,
    float* __restrict__ out)
{
    int p = blockIdx.x * 256 + threadIdx.x;
    if (p >= PLANE) return;
    int py = p / WW, px = p - py * WW;

    const float isg[8] = {0.31622776601683794f, 0.31622776601683794f,
                          7.0710678118654755f, 7.0710678118654755f, 7.0710678118654755f,
                          3.1622776601683795f, 3.1622776601683795f, 3.1622776601683795f};
    float cg[8], ce[3], cv[3];
#pragma unroll
    for (int c = 0; c < 8; ++c) cg[c] = guidance[c * PLANE + p] * isg[c];
#pragma unroll
    for (int c = 0; c < 3; ++c) { ce[c] = est[c * PLANE + p]; cv[c] = var[c * PLANE + p]; }

    float accx = 0.f, accy = 0.f, accz = 0.f, accw = 0.f;
    for (int dy = 0; dy < KW; ++dy) {
        const int sy = reflect_idx(py + dy - RADIUS, HH);
        for (int dx = 0; dx < KW; ++dx) {
            const int q = sy * WW + reflect_idx(px + dx - RADIUS, WW);
            float maha = 0.f;
#pragma unroll
            for (int c = 0; c < 8; ++c) {
                float d = guidance[c * PLANE + q] * isg[c] - cg[c];
                maha = fmaf(d, d, maha);
            }
            const float bil = fast_exp2(NEGHL2E * maha);
            int ok = 1;
#pragma unroll
            for (int c = 0; c < 3; ++c) {
                const float vq = var[c * PLANE + q];
                ok &= ttest_pass(ce[c] - est[c * PLANE + q], cv[c] + vq, cv[c], vq);
            }
            const float wgt = ok ? bil : 0.f;
            accx = fmaf(wgt, image[0 * PLANE + q], accx);
            accy = fmaf(wgt, image[1 * PLANE + q], accy);
            accz = fmaf(wgt, image[2 * PLANE + q], accz);
            accw += wgt;
        }
    }
    const float inv = 1.0f / fmaxf(accw, 1e-10f);
    out[p]             = accx * inv;
    out[p + PLANE]     = accy * inv;
    out[p + 2 * PLANE] = accz * inv;
}

// ---------------------------------------------------------------------------
extern "C" void kernel_launch(void* const* d_in, const int* in_sizes, int n_in,
                              void* d_out, int out_size, void* d_ws, size_t ws_size,
                              hipStream_t stream)
{
    const float* image    = (const float*)d_in[0];
    const float* guidance = (const float*)d_in[1];
    const float* est      = (const float*)d_in[2];
    const float* var      = (const float*)d_in[3];
    float* out = (float*)d_out;

    const size_t need = (size_t)NPIX * 20 * sizeof(float);   // ~5.66 MB
    if (ws_size >= need) {
        float* wsG  = (float*)d_ws;
        float* wsEV = wsG  + (size_t)NPIX * 8;
        float* wsI  = wsEV + (size_t)NPIX * 8;
        pack_pad<<<(NPIX + 255) / 256, 256, 0, stream>>>(image, guidance, est, var,
                                                         wsG, wsEV, wsI);
        dim3 grid(WW / TILE, HH / TILE);
        denoise_tiled<<<grid, 256, 0, stream>>>(wsG, wsEV, wsI, out);
    } else {
        denoise_direct<<<(PLANE + 255) / 256, 256, 0, stream>>>(image, guidance, est,
                                                                var, out);
    }
}